// DirectedODRLoss_36945308680520
// MI455X (gfx1250) — compile-verified
//
#include <hip/hip_runtime.h>
#include <hip/hip_bf16.h>

#define BN 4096
#define DK 256
#define KNN 25
#define FEPS 1e-8f

typedef __attribute__((ext_vector_type(2))) float v2f;
typedef __attribute__((ext_vector_type(8))) float v8f;

__global__ void init_loss_kernel(float* out) {
    if (threadIdx.x == 0 && blockIdx.x == 0) out[0] = 0.f;
}

__global__ void zero_u32_kernel(unsigned* p, int n) {
    int i = blockIdx.x * blockDim.x + threadIdx.x;
    if (i < n) p[i] = 0u;
}

// sq[i] = sum_d F[i][d]^2   (one 256-thread block per row, D=256)
__global__ __launch_bounds__(256)
void rownorm_kernel(const float* __restrict__ F, float* __restrict__ sq) {
    __shared__ float red[256];
    int row = blockIdx.x;
    float v = F[(size_t)row * DK + threadIdx.x];
    red[threadIdx.x] = v * v;
    __syncthreads();
    for (int s = 128; s > 0; s >>= 1) {
        if (threadIdx.x < s) red[threadIdx.x] += red[threadIdx.x + s];
        __syncthreads();
    }
    if (threadIdx.x == 0) sq[row] = red[0];
}

// C[M,N] = A[M,K] * B   where TRANSB: B is [N,K] row-major (multiply by B^T)
//                       else:        B is [K,N] row-major
// Each wave: 2x2 grid of 16x16 f32 WMMA tiles (32x32). Block = 8 waves = 64x128.
// Optional fused epilogue: loss += coef * sum(C .* relu(s_m - s_n)) without storing C.
template <bool TRANSB, bool STORE, bool FUSE>
__global__ __launch_bounds__(256)
void gemm_wmma_kernel(const float* __restrict__ A, const float* __restrict__ Bm,
                      float* __restrict__ C, int N, int K,
                      const float* __restrict__ scores, float coef,
                      float* __restrict__ loss)
{
    __shared__ float red[256];
    const int lane = threadIdx.x & 31;
    const int wave = threadIdx.x >> 5;
    const int l15  = lane & 15;
    const int lh   = lane >> 4;
    const int m0 = blockIdx.y * 64  + (wave >> 2) * 32;
    const int n0 = blockIdx.x * 128 + (wave & 3) * 32;

    v8f acc[2][2] = {};

    for (int k0 = 0; k0 < K; k0 += 4) {
        const int ka = k0 + lh * 2;     // lanes 0-15: K=k0,k0+1 ; lanes 16-31: K=k0+2,k0+3
        v2f a[2], b[2];
#pragma unroll
        for (int t = 0; t < 2; t++)
            a[t] = *(const v2f*)(A + (size_t)(m0 + t * 16 + l15) * K + ka);
        if (TRANSB) {
#pragma unroll
            for (int t = 0; t < 2; t++)
                b[t] = *(const v2f*)(Bm + (size_t)(n0 + t * 16 + l15) * K + ka);
        } else {
#pragma unroll
            for (int t = 0; t < 2; t++) {
                const int n = n0 + t * 16 + l15;
                b[t].x = Bm[(size_t)ka * N + n];
                b[t].y = Bm[(size_t)(ka + 1) * N + n];
            }
        }
#pragma unroll
        for (int i = 0; i < 2; i++)
#pragma unroll
            for (int j = 0; j < 2; j++)
                acc[i][j] = __builtin_amdgcn_wmma_f32_16x16x4_f32(
                    false, a[i], false, b[j], (short)0, acc[i][j], false, false);
    }

    float lsum = 0.f;
#pragma unroll
    for (int i = 0; i < 2; i++) {
#pragma unroll
        for (int j = 0; j < 2; j++) {
#pragma unroll
            for (int e = 0; e < 8; e++) {
                const int m = m0 + i * 16 + lh * 8 + e;
                const int n = n0 + j * 16 + l15;
                const float v = acc[i][j][e];
                if (STORE) C[(size_t)m * N + n] = v;
                if (FUSE) {
                    float p = scores[m] - scores[n];
                    lsum += v * (p > 0.f ? p : 0.f);
                }
            }
        }
    }
    if (FUSE) {
        red[threadIdx.x] = lsum;
        __syncthreads();
        for (int s = 128; s > 0; s >>= 1) {
            if (threadIdx.x < s) red[threadIdx.x] += red[threadIdx.x + s];
            __syncthreads();
        }
        if (threadIdx.x == 0) atomicAdd(loss, coef * red[0]);
    }
}

// In-place: G row -> clamped dist row; 25x block-argmin for kNN + sigma.
__global__ __launch_bounds__(256)
void dist_topk_kernel(float* __restrict__ G, const float* __restrict__ sq,
                      float* __restrict__ sigma, int* __restrict__ knn_idx)
{
    __shared__ float row[BN];
    __shared__ float rv[256];
    __shared__ int   ri[256];
    const int i = blockIdx.x;
    const float sqi = sq[i];
    for (int j = threadIdx.x; j < BN; j += 256) {
        float d = sqi + sq[j] - 2.f * G[(size_t)i * BN + j];
        d = d > 0.f ? d : 0.f;
        row[j] = d;
        G[(size_t)i * BN + j] = d;
    }
    __syncthreads();
    float acc = 0.f;
    for (int it = 0; it < KNN; it++) {
        float bv = 3.4e38f; int bi = 0;
        for (int j = threadIdx.x; j < BN; j += 256) {
            float v = row[j];
            if (v < bv) { bv = v; bi = j; }
        }
        rv[threadIdx.x] = bv; ri[threadIdx.x] = bi;
        __syncthreads();
        for (int s = 128; s > 0; s >>= 1) {
            if (threadIdx.x < s && rv[threadIdx.x + s] < rv[threadIdx.x]) {
                rv[threadIdx.x] = rv[threadIdx.x + s];
                ri[threadIdx.x] = ri[threadIdx.x + s];
            }
            __syncthreads();
        }
        const int widx = ri[0];
        const float wval = rv[0];
        __syncthreads();
        if (threadIdx.x == 0) {
            knn_idx[i * KNN + it] = widx;
            row[widx] = 3.4e38f;
        }
        acc += sqrtf(wval + FEPS);
        __syncthreads();
    }
    if (threadIdx.x == 0) sigma[i] = acc / (float)KNN;
}

__global__ void scatter_knn_kernel(const int* __restrict__ knn_idx, unsigned* __restrict__ mask)
{
    int t = blockIdx.x * blockDim.x + threadIdx.x;
    if (t >= BN * KNN) return;
    int i = t / KNN;
    int j = knn_idx[t];
    atomicOr(&mask[(size_t)i * (BN / 32) + (j >> 5)], 1u << (j & 31));
}

// dist row -> row of P (in place): W = exp(-d/(si*sj+eps))*mutual*(li<=lj) + I, normalize.
__global__ __launch_bounds__(256)
void build_P_kernel(float* __restrict__ D, const float* __restrict__ sigma,
                    const unsigned* __restrict__ mask, const int* __restrict__ labels)
{
    __shared__ float row[BN];
    __shared__ float red[256];
    const int i = blockIdx.x;
    const float si = sigma[i];
    const int li = labels[i];
    float lsum = 0.f;
    for (int j = threadIdx.x; j < BN; j += 256) {
        unsigned mij = (mask[(size_t)i * (BN / 32) + (j >> 5)] >> (j & 31)) & 1u;
        unsigned mji = (mask[(size_t)j * (BN / 32) + (i >> 5)] >> (i & 31)) & 1u;
        float w = 0.f;
        if ((mij & mji) && (li <= labels[j])) {
            float d = D[(size_t)i * BN + j];
            w = __expf(-d / (si * sigma[j] + FEPS));
        }
        if (j == i) w += 1.f;
        row[j] = w;
        lsum += w;
    }
    red[threadIdx.x] = lsum;
    __syncthreads();
    for (int s = 128; s > 0; s >>= 1) {
        if (threadIdx.x < s) red[threadIdx.x] += red[threadIdx.x + s];
        __syncthreads();
    }
    const float inv = 1.f / (red[0] + FEPS);
    for (int j = threadIdx.x; j < BN; j += 256)
        D[(size_t)i * BN + j] = row[j] * inv;
}

// loss += coef * sum_ij P_ij * relu(s_i - s_j)   (used for t=1)
__global__ __launch_bounds__(256)
void dot_penalty_kernel(const float* __restrict__ P, const float* __restrict__ s,
                        float coef, float* __restrict__ loss)
{
    __shared__ float red[256];
    const int i = blockIdx.x;
    const float si = s[i];
    float acc = 0.f;
    for (int j = threadIdx.x; j < BN; j += 256) {
        float p = si - s[j];
        acc += P[(size_t)i * BN + j] * (p > 0.f ? p : 0.f);
    }
    red[threadIdx.x] = acc;
    __syncthreads();
    for (int s2 = 128; s2 > 0; s2 >>= 1) {
        if (threadIdx.x < s2) red[threadIdx.x] += red[threadIdx.x + s2];
        __syncthreads();
    }
    if (threadIdx.x == 0) atomicAdd(loss, coef * red[0]);
}

extern "C" void kernel_launch(void* const* d_in, const int* in_sizes, int n_in,
                              void* d_out, int out_size, void* d_ws, size_t ws_size,
                              hipStream_t stream)
{
    const float* features = (const float*)d_in[0];
    const float* scores   = (const float*)d_in[1];
    const int*   labels   = (const int*)d_in[2];
    float* loss = (float*)d_out;

    char* ws = (char*)d_ws;
    const size_t MAT = (size_t)BN * BN * sizeof(float);   // 64 MB
    float*    bufA  = (float*)ws;                 // G -> dist -> P (in place)
    float*    bufB  = (float*)(ws + MAT);         // P^2
    float*    sq    = (float*)(ws + 2 * MAT);
    float*    sigma = sq + BN;
    int*      knn   = (int*)(sigma + BN);
    unsigned* mask  = (unsigned*)(knn + BN * KNN);

    init_loss_kernel<<<1, 64, 0, stream>>>(loss);
    zero_u32_kernel<<<(BN * (BN / 32) + 255) / 256, 256, 0, stream>>>(mask, BN * (BN / 32));
    rownorm_kernel<<<BN, 256, 0, stream>>>(features, sq);

    const dim3 gb(BN / 128, BN / 64);
    // G = F * F^T  (f32 WMMA, K=256)
    gemm_wmma_kernel<true, true, false><<<gb, 256, 0, stream>>>(
        features, features, bufA, BN, DK, nullptr, 0.f, nullptr);

    dist_topk_kernel<<<BN, 256, 0, stream>>>(bufA, sq, sigma, knn);
    scatter_knn_kernel<<<(BN * KNN + 255) / 256, 256, 0, stream>>>(knn, mask);
    build_P_kernel<<<BN, 256, 0, stream>>>(bufA, sigma, mask, labels);

    const float invB = 1.f / (float)BN;
    // t=1: direct dot with penalty
    dot_penalty_kernel<<<BN, 256, 0, stream>>>(bufA, scores, invB, loss);
    // t=2: P2 = P @ P (store) + fused loss epilogue, coef = 1/(2B)
    gemm_wmma_kernel<false, true, true><<<gb, 256, 0, stream>>>(
        bufA, bufA, bufB, BN, BN, scores, 0.5f * invB, loss);
    // t=3: P3 = P2 @ P, never stored — fused loss only, coef = 1/(3B)
    gemm_wmma_kernel<false, false, true><<<gb, 256, 0, stream>>>(
        bufB, bufA, nullptr, BN, BN, scores, invB / 3.f, loss);
}